// GatedDeltaNet_26792005992797
// MI455X (gfx1250) — compile-verified
//
#include <hip/hip_runtime.h>
#include <hip/hip_bf16.h>
#include <math.h>
#include <stdint.h>

// Problem constants (from the reference)
#define Bc   2
#define Sc   4096
#define Hc   2048
#define NHc  8
#define Dc   128
#define KDc  1024           // NH*D
#define Wc   4
#define ROWS (Bc*Sc)        // 8192
#define LOG_ROPE_BASE 13.815510557964274f   // ln(1e6)

typedef float v2f __attribute__((ext_vector_type(2)));
typedef float v8f __attribute__((ext_vector_type(8)));

// ---------------------------------------------------------------------------
// CDNA5 async global->LDS copy (ASYNCcnt-tracked).
// The ROCm clang builtin for this op has a descriptor-style signature (its
// first param is an i32x4), so per the CDNA5 bridge-doc guidance we emit the
// ISA instruction directly with inline asm: VGLOBAL encoding, VDST = LDS byte
// offset VGPR (low 32 bits of a generic __shared__ pointer), VADDR = 64-bit
// global address pair, no SADDR.
// ---------------------------------------------------------------------------
static __device__ __forceinline__ void async_copy_b128(const float* g, float* l)
{
    unsigned int loff = (unsigned int)(uintptr_t)l;   // low 32 bits = LDS offset
    asm volatile("global_load_async_to_lds_b128 %0, %1, off"
                 :: "v"(loff), "v"(g) : "memory");
}

#if defined(__has_builtin)
#if __has_builtin(__builtin_amdgcn_s_wait_asynccnt)
#define HAVE_ASYNC_WAIT_BUILTIN 1
#endif
#endif

#if defined(HAVE_ASYNC_WAIT_BUILTIN)
#define WAIT_ASYNC(n) __builtin_amdgcn_s_wait_asynccnt(n)
#else
#define WAIT_ASYNC(n) asm volatile("s_wait_asynccnt %0" :: "n"(n))
#endif

// ---------------------------------------------------------------------------
// FP32 WMMA GEMM: C[M,N] = A[M,K] @ B[K,N]
// Block = 256 threads = 8 waves. Block tile 128x64, BK=16, double-buffered LDS
// filled by async global->LDS B128 copies. Wave w owns a 32x32 sub-tile
// (2x2 accumulators of 16x16). All call sites: M%128==0, N%64==0, K%16==0.
// ---------------------------------------------------------------------------
__global__ __launch_bounds__(256) void wmma_gemm_f32(const float* __restrict__ A,
                                                     const float* __restrict__ B,
                                                     float* __restrict__ C,
                                                     int M, int N, int K)
{
    // A rows padded to 20 dwords: conflict-free (20*m16 mod 64 distinct for
    // m16=0..15) and 16B-aligned for async B128 writes.
    __shared__ float As[2][128][20];
    __shared__ float Bs[2][16][64];

    const int t    = threadIdx.x;
    const int wave = t >> 5;
    const int lane = t & 31;

    const int bm = blockIdx.y * 128;
    const int bn = blockIdx.x * 64;

    const int wr = (wave & 3) * 32;       // wave row base within block tile
    const int wc = (wave >> 2) * 32;      // wave col base within block tile

    // async-stage assignments (one B128 per thread per target tile)
    const int la_row = t >> 2;            // 0..63  (A rows r and r+64)
    const int la_col = (t & 3) * 4;       // 0,4,8,12
    const int lb_row = t >> 4;            // 0..15
    const int lb_col = (t & 15) * 4;      // 0..60

    // WMMA f32 16x16x4 fragment addressing (wave32)
    const int m16 = lane & 15;
    const int kb  = (lane >> 4) * 2;      // lanes 0-15 -> K0/1, lanes 16-31 -> K2/3

    const int ar0 = wr + m16;             // A fragment rows
    const int ar1 = ar0 + 16;
    const int bc0 = wc + m16;             // B fragment cols
    const int bc1 = bc0 + 16;

    v8f acc00 = {}; v8f acc01 = {};
    v8f acc10 = {}; v8f acc11 = {};

    auto issue_stage = [&](int nb, int k0) {
        async_copy_b128(&A[(size_t)(bm + la_row) * K + (k0 + la_col)],
                        &As[nb][la_row][la_col]);
        async_copy_b128(&A[(size_t)(bm + 64 + la_row) * K + (k0 + la_col)],
                        &As[nb][64 + la_row][la_col]);
        async_copy_b128(&B[(size_t)(k0 + lb_row) * N + (bn + lb_col)],
                        &Bs[nb][lb_row][lb_col]);
    };

    issue_stage(0, 0);
    int buf = 0;

    for (int k0 = 0; k0 < K; k0 += 16) {
        const bool more = (k0 + 16) < K;
        if (more) {
            issue_stage(buf ^ 1, k0 + 16);   // overlap next fill with compute
            WAIT_ASYNC(3);                   // 3 pending = this wave's next-stage ops;
                                             // in-order completion => buf's data landed
        } else {
            WAIT_ASYNC(0);
        }
        __syncthreads();                     // all waves' tile data visible

#pragma unroll
        for (int ks = 0; ks < 16; ks += 4) {
            v2f a0, a1, b0, b1;
            a0.x = As[buf][ar0][ks + kb];  a0.y = As[buf][ar0][ks + kb + 1];
            a1.x = As[buf][ar1][ks + kb];  a1.y = As[buf][ar1][ks + kb + 1];
            b0.x = Bs[buf][ks + kb][bc0];  b0.y = Bs[buf][ks + kb + 1][bc0];
            b1.x = Bs[buf][ks + kb][bc1];  b1.y = Bs[buf][ks + kb + 1][bc1];

            acc00 = __builtin_amdgcn_wmma_f32_16x16x4_f32(false, a0, false, b0,
                                                          (short)0, acc00, false, false);
            acc01 = __builtin_amdgcn_wmma_f32_16x16x4_f32(false, a0, false, b1,
                                                          (short)0, acc01, false, false);
            acc10 = __builtin_amdgcn_wmma_f32_16x16x4_f32(false, a1, false, b0,
                                                          (short)0, acc10, false, false);
            acc11 = __builtin_amdgcn_wmma_f32_16x16x4_f32(false, a1, false, b1,
                                                          (short)0, acc11, false, false);
        }
        __syncthreads();                     // done reading buf; safe to refill next iter
        buf ^= 1;
    }

    // C/D layout: VGPR i -> M = base + i (lanes 0-15) or base + 8 + i (lanes 16-31)
    const int cm0 = bm + wr + 8 * (lane >> 4);
    const int cn0 = bn + wc + (lane & 15);
#pragma unroll
    for (int i = 0; i < 8; ++i) {
        C[(size_t)(cm0 + i) * N + cn0]           = acc00[i];
        C[(size_t)(cm0 + i) * N + cn0 + 16]      = acc01[i];
        C[(size_t)(cm0 + 16 + i) * N + cn0]      = acc10[i];
        C[(size_t)(cm0 + 16 + i) * N + cn0 + 16] = acc11[i];
    }
}

// ---------------------------------------------------------------------------
// bg = x @ W_b, ag = x @ W_a   (N = NH = 8, too narrow for WMMA tiles)
// One block per row; x row staged in LDS; wave w computes head-w dots.
// W_b/W_a are 64KB each -> L2-resident across the whole grid.
// ---------------------------------------------------------------------------
__global__ __launch_bounds__(256) void gates_kernel(const float* __restrict__ x,
                                                    const float* __restrict__ Wb,
                                                    const float* __restrict__ Wa,
                                                    float* __restrict__ bg,
                                                    float* __restrict__ ag)
{
    __shared__ float xs[Hc];
    const int row = blockIdx.x;
    const float* xr = x + (size_t)row * Hc;
    for (int i = threadIdx.x; i < Hc; i += 256) xs[i] = xr[i];
    __syncthreads();

    const int wave = threadIdx.x >> 5;
    const int lane = threadIdx.x & 31;

    float sb = 0.0f, sa = 0.0f;
    for (int i = lane; i < Hc; i += 32) {
        const float xv = xs[i];
        sb += xv * Wb[(size_t)i * NHc + wave];
        sa += xv * Wa[(size_t)i * NHc + wave];
    }
#pragma unroll
    for (int m = 16; m; m >>= 1) {
        sb += __shfl_xor(sb, m, 32);
        sa += __shfl_xor(sa, m, 32);
    }
    if (lane == 0) {
        bg[(size_t)row * NHc + wave] = sb;
        ag[(size_t)row * NHc + wave] = sa;
    }
}

// ---------------------------------------------------------------------------
// RoPE + 5-tap gated window attention + swish gate + RMSNorm -> yn
// One block per (b,s) row. Wave w == head w (32 lanes x 4 elems = D = 128).
// qkv rows are [q(1024) | k(1024) | v(1024)].
// ---------------------------------------------------------------------------
__device__ __forceinline__ float sigf(float v) { return 1.0f / (1.0f + __expf(-v)); }

__global__ __launch_bounds__(256) void attn_kernel(const float* __restrict__ qkv,
                                                   const float* __restrict__ z,
                                                   const float* __restrict__ bg,
                                                   const float* __restrict__ ag,
                                                   const float* __restrict__ norm_w,
                                                   float* __restrict__ yn)
{
    const int r    = blockIdx.x;          // 0..ROWS-1
    const int s    = r & (Sc - 1);        // S = 4096 (power of 2)
    const int base = r - s;               // b*S
    const int h    = threadIdx.x >> 5;    // head
    const int lane = threadIdx.x & 31;
    const int col  = h * Dc + lane * 4;   // column within KD

    const size_t qrow = (size_t)r * (3 * KDc);

    // RoPE inverse frequencies for this thread's two (even,odd) pairs
    const float f0 = __expf(-((2.0f * (float)(lane * 2    )) / (float)Dc) * LOG_ROPE_BASE);
    const float f1 = __expf(-((2.0f * (float)(lane * 2 + 1)) / (float)Dc) * LOG_ROPE_BASE);

    const float q0 = qkv[qrow + col], q1 = qkv[qrow + col + 1];
    const float q2 = qkv[qrow + col + 2], q3 = qkv[qrow + col + 3];

    float sn0, cs0, sn1, cs1;
    __sincosf((float)s * f0, &sn0, &cs0);
    __sincosf((float)s * f1, &sn1, &cs1);
    const float qr0 = q0 * cs0 - q1 * sn0, qr1 = q0 * sn0 + q1 * cs0;
    const float qr2 = q2 * cs1 - q3 * sn1, qr3 = q2 * sn1 + q3 * cs1;

    const float agv = ag[(size_t)r * NHc + h];

    float o0 = 0.0f, o1 = 0.0f, o2 = 0.0f, o3 = 0.0f;
    if (s == 0) {
        const size_t vrow = qrow + 2 * KDc;
        o0 = qkv[vrow + col];     o1 = qkv[vrow + col + 1];
        o2 = qkv[vrow + col + 2]; o3 = qkv[vrow + col + 3];
    } else {
        const float scale = 0.08838834764831845f;  // 1/sqrt(128)
#pragma unroll
        for (int off = 0; off <= Wc; ++off) {
            const int j = s - off;
            if (j < 0) break;
            const size_t jrow = (size_t)(base + j) * (3 * KDc);
            const float k0 = qkv[jrow + KDc + col],     k1 = qkv[jrow + KDc + col + 1];
            const float k2 = qkv[jrow + KDc + col + 2], k3 = qkv[jrow + KDc + col + 3];
            float snj0, csj0, snj1, csj1;
            __sincosf((float)j * f0, &snj0, &csj0);
            __sincosf((float)j * f1, &snj1, &csj1);
            const float kr0 = k0 * csj0 - k1 * snj0, kr1 = k0 * snj0 + k1 * csj0;
            const float kr2 = k2 * csj1 - k3 * snj1, kr3 = k2 * snj1 + k3 * csj1;

            float part = qr0 * kr0 + qr1 * kr1 + qr2 * kr2 + qr3 * kr3;
#pragma unroll
            for (int m = 16; m; m >>= 1) part += __shfl_xor(part, m, 32);

            const float score = part * scale;
            const float bgv   = bg[(size_t)(base + j) * NHc + h];
            const float sval  = score * sigf(agv * bgv);

            const float v0 = qkv[jrow + 2 * KDc + col],     v1 = qkv[jrow + 2 * KDc + col + 1];
            const float v2 = qkv[jrow + 2 * KDc + col + 2], v3 = qkv[jrow + 2 * KDc + col + 3];
            o0 += sval * v0; o1 += sval * v1; o2 += sval * v2; o3 += sval * v3;
        }
    }

    // y = out * sigmoid(z); RMS over the KD row; yn = y/rms * norm_w * swish(z)
    const size_t zrow = (size_t)r * KDc;
    const float z0 = z[zrow + col],     z1 = z[zrow + col + 1];
    const float z2 = z[zrow + col + 2], z3 = z[zrow + col + 3];
    const float y0 = o0 * sigf(z0), y1 = o1 * sigf(z1);
    const float y2 = o2 * sigf(z2), y3 = o3 * sigf(z3);

    float part = y0 * y0 + y1 * y1 + y2 * y2 + y3 * y3;
#pragma unroll
    for (int m = 16; m; m >>= 1) part += __shfl_xor(part, m, 32);

    __shared__ float red[NHc];
    if (lane == 0) red[h] = part;
    __syncthreads();
    float tot = 0.0f;
#pragma unroll
    for (int i = 0; i < NHc; ++i) tot += red[i];
    const float rinv = rsqrtf(tot / (float)KDc + 1e-6f);

    const float w0 = norm_w[col],     w1 = norm_w[col + 1];
    const float w2 = norm_w[col + 2], w3 = norm_w[col + 3];
    yn[zrow + col]     = y0 * rinv * w0 * (z0 * sigf(z0));
    yn[zrow + col + 1] = y1 * rinv * w1 * (z1 * sigf(z1));
    yn[zrow + col + 2] = y2 * rinv * w2 * (z2 * sigf(z2));
    yn[zrow + col + 3] = y3 * rinv * w3 * (z3 * sigf(z3));
}

// ---------------------------------------------------------------------------
extern "C" void kernel_launch(void* const* d_in, const int* in_sizes, int n_in,
                              void* d_out, int out_size, void* d_ws, size_t ws_size,
                              hipStream_t stream)
{
    const float* x      = (const float*)d_in[0];   // (B,S,H)
    const float* W_qkv  = (const float*)d_in[1];   // (H, 3KD)
    const float* W_z    = (const float*)d_in[2];   // (H, KD)
    const float* W_b    = (const float*)d_in[3];   // (H, NH)
    const float* W_a    = (const float*)d_in[4];   // (H, NH)
    const float* norm_w = (const float*)d_in[5];   // (KD,)
    const float* W_out  = (const float*)d_in[6];   // (KD, H)
    float* out = (float*)d_out;

    // workspace carve-up (floats)
    float* ws   = (float*)d_ws;
    float* qkv  = ws;                                   // ROWS * 3KD
    float* zb   = qkv + (size_t)ROWS * (3 * KDc);       // ROWS * KD
    float* bgp  = zb  + (size_t)ROWS * KDc;             // ROWS * NH
    float* agp  = bgp + (size_t)ROWS * NHc;             // ROWS * NH
    float* ynp  = agp + (size_t)ROWS * NHc;             // ROWS * KD

    dim3 blk(256);

    // qkv = x @ W_qkv   (M=8192, N=3072, K=2048)
    wmma_gemm_f32<<<dim3(3 * KDc / 64, ROWS / 128), blk, 0, stream>>>(
        x, W_qkv, qkv, ROWS, 3 * KDc, Hc);
    // z = x @ W_z       (M=8192, N=1024, K=2048)
    wmma_gemm_f32<<<dim3(KDc / 64, ROWS / 128), blk, 0, stream>>>(
        x, W_z, zb, ROWS, KDc, Hc);
    // bg, ag            (N=8 each)
    gates_kernel<<<ROWS, blk, 0, stream>>>(x, W_b, W_a, bgp, agp);
    // RoPE + windowed attention + gating + RMSNorm -> yn
    attn_kernel<<<ROWS, blk, 0, stream>>>(qkv, zb, bgp, agp, norm_w, ynp);
    // out = yn @ W_out  (M=8192, N=2048, K=1024)
    wmma_gemm_f32<<<dim3(Hc / 64, ROWS / 128), blk, 0, stream>>>(
        ynp, W_out, out, ROWS, Hc, KDc);
}